// SAGEConvGNN_28020366639512
// MI455X (gfx1250) — compile-verified
//
#include <hip/hip_runtime.h>

#define NN 20000
#define NE 640000
#define HID 256

typedef __attribute__((ext_vector_type(2))) float v2f;
typedef __attribute__((ext_vector_type(8))) float v8f;

__device__ __forceinline__ v8f wmma_f32_16x16x4(v2f a, v2f b, v8f c) {
  // 8 args: (neg_a, A, neg_b, B, c_mod, C, reuse_a, reuse_b)
  return __builtin_amdgcn_wmma_f32_16x16x4_f32(false, a, false, b, (short)0, c,
                                               false, false);
}

// ---------------------------------------------------------------- features
__global__ __launch_bounds__(256) void build_features(
    const float* __restrict__ op_feats, const float* __restrict__ cfg,
    const float* __restrict__ embed, const float* __restrict__ fmean,
    const float* __restrict__ fstd, const int* __restrict__ op_ids,
    float* __restrict__ x) {
  const int node = blockIdx.x;
  const int c = threadIdx.x;  // 0..255
  float v;
  if (c < 140)
    v = op_feats[node * 140 + c];
  else if (c < 204)
    v = embed[op_ids[node] * 64 + (c - 140)];
  else
    v = cfg[node * 52 + (c - 204)];
  x[node * HID + c] = (v - fmean[c]) / fstd[c];
}

// ---------------------------------------------------------------- degree/CSR
__global__ __launch_bounds__(256) void zero_i32(int* __restrict__ p, int n) {
  int i = blockIdx.x * 256 + threadIdx.x;
  if (i < n) p[i] = 0;
}

__global__ __launch_bounds__(256) void count_deg(const int* __restrict__ dst,
                                                 int* __restrict__ degi) {
  int e = blockIdx.x * 256 + threadIdx.x;
  if (e < NE) atomicAdd(&degi[dst[e]], 1);
}

__global__ __launch_bounds__(256) void inv_deg_k(const int* __restrict__ degi,
                                                 float* __restrict__ invdeg) {
  int i = blockIdx.x * 256 + threadIdx.x;
  if (i < NN) invdeg[i] = 1.0f / (float)max(degi[i], 1);
}

// Exclusive prefix sum over degi -> rowstart, three tiny kernels.
__global__ __launch_bounds__(256) void scan_block(const int* __restrict__ degi,
                                                  int* __restrict__ exc,
                                                  int* __restrict__ bsum) {
  __shared__ int sh[256];
  const int i = blockIdx.x * 256 + threadIdx.x;
  const int v = (i < NN) ? degi[i] : 0;
  sh[threadIdx.x] = v;
  __syncthreads();
  for (int off = 1; off < 256; off <<= 1) {
    int t = (threadIdx.x >= off) ? sh[threadIdx.x - off] : 0;
    __syncthreads();
    sh[threadIdx.x] += t;
    __syncthreads();
  }
  if (i < NN) exc[i] = sh[threadIdx.x] - v;
  if (threadIdx.x == 255) bsum[blockIdx.x] = sh[255];
}

__global__ __launch_bounds__(256) void scan_bsums(int* __restrict__ bsum,
                                                  int nblocks) {
  __shared__ int sh[256];
  const int v = (threadIdx.x < nblocks) ? bsum[threadIdx.x] : 0;
  sh[threadIdx.x] = v;
  __syncthreads();
  for (int off = 1; off < 256; off <<= 1) {
    int t = (threadIdx.x >= off) ? sh[threadIdx.x - off] : 0;
    __syncthreads();
    sh[threadIdx.x] += t;
    __syncthreads();
  }
  if (threadIdx.x < nblocks) bsum[threadIdx.x] = sh[threadIdx.x] - v;
}

__global__ __launch_bounds__(256) void scan_add(int* __restrict__ exc,
                                                const int* __restrict__ bsum,
                                                int* __restrict__ cursor) {
  const int i = blockIdx.x * 256 + threadIdx.x;
  if (i < NN) {
    const int r = exc[i] + bsum[blockIdx.x];
    exc[i] = r;       // rowstart
    cursor[i] = r;    // running fill cursor
  }
}

__global__ __launch_bounds__(256) void build_csr(const int* __restrict__ src,
                                                 const int* __restrict__ dst,
                                                 int* __restrict__ cursor,
                                                 int* __restrict__ edge_src) {
  const int e = blockIdx.x * 256 + threadIdx.x;
  if (e < NE) {
    const int pos = atomicAdd(&cursor[dst[e]], 1);
    edge_src[pos] = src[e];
  }
}

// ------------------------------------------------------- gather aggregation
// One block per node; 256 lanes each own one feature column. Pure L2-resident
// row gathers, no atomics. Folds the 1/deg mean scaling.
__global__ __launch_bounds__(256) void aggregate_mean(
    const float* __restrict__ x, const int* __restrict__ edge_src,
    const int* __restrict__ rowstart, const int* __restrict__ degi,
    const float* __restrict__ invdeg, float* __restrict__ agg) {
  const int node = blockIdx.x;
  const int c = threadIdx.x;
  const int beg = rowstart[node];
  const int n = degi[node];
  float a0 = 0.0f, a1 = 0.0f, a2 = 0.0f, a3 = 0.0f;
  int k = 0;
  for (; k + 4 <= n; k += 4) {
    const int s0 = edge_src[beg + k + 0];
    const int s1 = edge_src[beg + k + 1];
    const int s2 = edge_src[beg + k + 2];
    const int s3 = edge_src[beg + k + 3];
    if (k + 8 < n) {  // prefetch two iterations ahead (global_prefetch_b8)
      __builtin_prefetch(x + (size_t)edge_src[beg + k + 8] * HID + c, 0, 0);
    }
    a0 += x[(size_t)s0 * HID + c];
    a1 += x[(size_t)s1 * HID + c];
    a2 += x[(size_t)s2 * HID + c];
    a3 += x[(size_t)s3 * HID + c];
  }
  for (; k < n; ++k) a0 += x[(size_t)edge_src[beg + k] * HID + c];
  agg[(size_t)node * HID + c] = ((a0 + a1) + (a2 + a3)) * invdeg[node];
}

// ---------------------------------------------------------------- fused GEMM
// out[16, NOUT] = relu(LN( xin_tile @ Wa (+ agg_tile @ Wb) + bias ))
// Block: 256 threads = 8 waves; one 16-row node tile per block.
template <int NOUT, bool HAS_NEIGH>
__global__ __launch_bounds__(256) void gemm_ln_relu(
    const float* __restrict__ xin, const float* __restrict__ agg,
    const float* __restrict__ Wa, const float* __restrict__ Wb,
    const float* __restrict__ bias, const float* __restrict__ gamma,
    const float* __restrict__ beta, float* __restrict__ out) {
  constexpr int NT = NOUT / 128;  // 2 for NOUT=256, 1 for NOUT=128
  __shared__ float xs[16 * 257];  // padded stride: conflict-free column reads
  __shared__ float gs[HAS_NEIGH ? 16 * 257 : 16];
  __shared__ float hb[16 * (NOUT + 1)];

  const int tid = threadIdx.x;
  const int m0 = blockIdx.x * 16;

  for (int j = 0; j < 16; ++j) {
    xs[j * 257 + tid] = xin[(size_t)(m0 + j) * HID + tid];
    if constexpr (HAS_NEIGH) {
      gs[j * 257 + tid] = agg[(size_t)(m0 + j) * HID + tid];
    }
  }
  __syncthreads();

  const int wave = tid >> 5;
  const int lane = tid & 31;
  const int halfk = (lane >> 4) & 1;  // K sub-pair per ISA A/B layout
  const int lrow = lane & 15;         // M for A, N for B

  v8f acc[NT] = {};

  for (int k0 = 0; k0 < HID; k0 += 4) {
    const int ka = k0 + 2 * halfk;
    v2f ax;
    ax.x = xs[lrow * 257 + ka];
    ax.y = xs[lrow * 257 + ka + 1];
    v2f ag;
    if constexpr (HAS_NEIGH) {
      ag.x = gs[lrow * 257 + ka];
      ag.y = gs[lrow * 257 + ka + 1];
    }
#pragma unroll
    for (int i = 0; i < NT; ++i) {
      const int col = (wave * NT + i) * 16 + lrow;
      v2f bx;
      bx.x = Wa[(size_t)ka * NOUT + col];
      bx.y = Wa[(size_t)(ka + 1) * NOUT + col];
      acc[i] = wmma_f32_16x16x4(ax, bx, acc[i]);
      if constexpr (HAS_NEIGH) {
        v2f bg;
        bg.x = Wb[(size_t)ka * NOUT + col];
        bg.y = Wb[(size_t)(ka + 1) * NOUT + col];
        acc[i] = wmma_f32_16x16x4(ag, bg, acc[i]);
      }
    }
  }

#pragma unroll
  for (int i = 0; i < NT; ++i) {
    const int col = (wave * NT + i) * 16 + lrow;
    const float bcol = bias ? bias[col] : 0.0f;
#pragma unroll
    for (int r = 0; r < 8; ++r) {
      const int row = r + 8 * halfk;  // C layout: VGPR r -> M=r / M=r+8
      hb[row * (NOUT + 1) + col] = acc[i][r] + bcol;
    }
  }
  __syncthreads();

  // LayerNorm + ReLU: 16 lanes per row; xor-shuffles (mask<16) reduce within
  // each wave32 half independently.
  const int row = tid >> 4;
  const int sub = tid & 15;
  float s = 0.0f, ss = 0.0f;
#pragma unroll
  for (int j = 0; j < NOUT / 16; ++j) {
    const float v = hb[row * (NOUT + 1) + sub + 16 * j];
    s += v;
    ss += v * v;
  }
#pragma unroll
  for (int m = 8; m >= 1; m >>= 1) {
    s += __shfl_xor(s, m, 32);
    ss += __shfl_xor(ss, m, 32);
  }
  const float mu = s * (1.0f / NOUT);
  const float var = ss * (1.0f / NOUT) - mu * mu;
  const float rstd = rsqrtf(var + 1e-5f);
#pragma unroll
  for (int j = 0; j < NOUT / 16; ++j) {
    const int col = sub + 16 * j;
    const float v = hb[row * (NOUT + 1) + col];
    const float y = (v - mu) * rstd * gamma[col] + beta[col];
    out[(size_t)(m0 + row) * NOUT + col] = fmaxf(y, 0.0f);
  }
}

// ---------------------------------------------------------------- launch
extern "C" void kernel_launch(void* const* d_in, const int* in_sizes, int n_in,
                              void* d_out, int out_size, void* d_ws,
                              size_t ws_size, hipStream_t stream) {
  (void)in_sizes; (void)n_in; (void)out_size; (void)ws_size;

  const float* op_feats = (const float*)d_in[0];
  const float* cfg      = (const float*)d_in[1];
  const float* embed    = (const float*)d_in[2];
  const float* fmean    = (const float*)d_in[3];
  const float* fstd     = (const float*)d_in[4];
  const float* Wself    = (const float*)d_in[5];   // [4,256,256]
  const float* Wneigh   = (const float*)d_in[6];   // [4,256,256]
  const float* bconv    = (const float*)d_in[7];   // [4,256]
  const float* cgamma   = (const float*)d_in[8];   // [4,256]
  const float* cbeta    = (const float*)d_in[9];   // [4,256]
  const float* W1       = (const float*)d_in[10];  // [256,256]
  const float* g1       = (const float*)d_in[11];
  const float* b1       = (const float*)d_in[12];
  const float* W2       = (const float*)d_in[13];  // [256,128]
  const float* g2       = (const float*)d_in[14];
  const float* b2       = (const float*)d_in[15];
  const int* op_ids     = (const int*)d_in[16];
  const int* src        = (const int*)d_in[17];
  const int* dst        = (const int*)d_in[18];

  float* ws = (float*)d_ws;
  float* xA = ws;
  float* xB = xA + (size_t)NN * HID;
  float* agg = xB + (size_t)NN * HID;
  float* invdeg = agg + (size_t)NN * HID;
  int* degi = (int*)(invdeg + NN);
  int* rowstart = degi + NN;
  int* cursor = rowstart + NN;
  int* bsum = cursor + NN;
  int* edge_src = bsum + 256;

  const int nb_nodes = (NN + 255) / 256;  // 79 (<=256, fits one scan block)
  const int nb_edges = NE / 256;          // 2500

  build_features<<<NN, 256, 0, stream>>>(op_feats, cfg, embed, fmean, fstd,
                                         op_ids, xA);

  // Build CSR-by-destination once; reused by all 4 layers.
  zero_i32<<<nb_nodes, 256, 0, stream>>>(degi, NN);
  count_deg<<<nb_edges, 256, 0, stream>>>(dst, degi);
  inv_deg_k<<<nb_nodes, 256, 0, stream>>>(degi, invdeg);
  scan_block<<<nb_nodes, 256, 0, stream>>>(degi, rowstart, bsum);
  scan_bsums<<<1, 256, 0, stream>>>(bsum, nb_nodes);
  scan_add<<<nb_nodes, 256, 0, stream>>>(rowstart, bsum, cursor);
  build_csr<<<nb_edges, 256, 0, stream>>>(src, dst, cursor, edge_src);

  float* xcur = xA;
  float* xnext = xB;
  for (int l = 0; l < 4; ++l) {
    aggregate_mean<<<NN, 256, 0, stream>>>(xcur, edge_src, rowstart, degi,
                                           invdeg, agg);
    gemm_ln_relu<256, true><<<NN / 16, 256, 0, stream>>>(
        xcur, agg, Wself + (size_t)l * HID * HID,
        Wneigh + (size_t)l * HID * HID, bconv + l * HID, cgamma + l * HID,
        cbeta + l * HID, xnext);
    float* t = xcur; xcur = xnext; xnext = t;
  }

  // Head: Dense(256) -> LN -> ReLU -> Dense(128) -> LN -> ReLU
  gemm_ln_relu<256, false><<<NN / 16, 256, 0, stream>>>(
      xcur, nullptr, W1, nullptr, nullptr, g1, b1, xnext);
  gemm_ln_relu<128, false><<<NN / 16, 256, 0, stream>>>(
      xnext, nullptr, W2, nullptr, nullptr, g2, b2, (float*)d_out);
}